// Net_26027501814170
// MI455X (gfx1250) — compile-verified
//
#include <hip/hip_runtime.h>

// LIF SNN, 2 layers x 2 channels, 100 serial steps, B = 131072.
// Memory-bound: ~210 MB of streaming f32 output -> use b128 non-temporal
// stores (gfx1250 TH_STORE_NT). WMMA does not apply (2x2 matvecs on a
// store-bandwidth-bound problem). One thread = 2 batch elements (4 channels),
// so each per-step record (spk2 quad, mem2 quad) is one float4 store.
//
// Key identity: reset_{t+1} = (mem_t - 1 > 0) = spk_t, so the spike value is
// carried as next step's reset (mem0 = 0 -> reset_0 = 0 matches).

#define SNN_BETA   0.99f
#define SNN_THRESH 1.0f
#define SNN_STEPS  100

typedef __attribute__((ext_vector_type(4))) float v4f;

__global__ __launch_bounds__(256) void snn_lif_kernel(
    const float* __restrict__ x,   // [B,2]
    const float* __restrict__ W1,  // [2,2] row-major
    const float* __restrict__ b1,  // [2]
    const float* __restrict__ W2,  // [2,2] row-major
    const float* __restrict__ b2,  // [2]
    float* __restrict__ out,       // spk2_rec [100,B,2] ++ mem2_rec [100,B,2]
    int B)
{
    const int tid = blockIdx.x * blockDim.x + threadIdx.x; // owns batch 2*tid, 2*tid+1
    if (2 * tid >= B) return;

    // Uniform params -> SMEM scalar loads.
    const float w1_00 = W1[0], w1_01 = W1[1], w1_10 = W1[2], w1_11 = W1[3];
    const float b1_0  = b1[0], b1_1  = b1[1];
    const float w2_00 = W2[0], w2_01 = W2[1], w2_10 = W2[2], w2_11 = W2[3];
    const float b2_0  = b2[0], b2_1  = b2[1];

    // x for both elements: 4 contiguous floats (b128 load).
    const v4f xv = reinterpret_cast<const v4f*>(x)[tid];

    // cur1[b,c] = b1[c] + x[b,0]*W1[c,0] + x[b,1]*W1[c,1]  (constant over steps)
    const float c1a0 = fmaf(xv.x, w1_00, fmaf(xv.y, w1_01, b1_0));
    const float c1a1 = fmaf(xv.x, w1_10, fmaf(xv.y, w1_11, b1_1));
    const float c1b0 = fmaf(xv.z, w1_00, fmaf(xv.w, w1_01, b1_0));
    const float c1b1 = fmaf(xv.z, w1_10, fmaf(xv.w, w1_11, b1_1));

    // State: membranes and spikes (spike doubles as next-step reset).
    float m1a0 = 0.f, m1a1 = 0.f, m1b0 = 0.f, m1b1 = 0.f;
    float m2a0 = 0.f, m2a1 = 0.f, m2b0 = 0.f, m2b1 = 0.f;
    float s1a0 = 0.f, s1a1 = 0.f, s1b0 = 0.f, s1b1 = 0.f;
    float s2a0 = 0.f, s2a1 = 0.f, s2b0 = 0.f, s2b1 = 0.f;

    const size_t step = (size_t)B * 2;                       // floats per [B,2] slab
    float* spkPtr = out + (size_t)tid * 4;                   // spk2_rec block
    float* memPtr = out + (size_t)SNN_STEPS * step + (size_t)tid * 4; // mem2_rec block

    #pragma unroll 4
    for (int t = 0; t < SNN_STEPS; ++t) {
        // ---- layer 1: mem = beta*mem + cur1 - spk_prev ; spk = (mem > 1)
        m1a0 = fmaf(SNN_BETA, m1a0, c1a0) - s1a0;
        m1a1 = fmaf(SNN_BETA, m1a1, c1a1) - s1a1;
        m1b0 = fmaf(SNN_BETA, m1b0, c1b0) - s1b0;
        m1b1 = fmaf(SNN_BETA, m1b1, c1b1) - s1b1;
        s1a0 = (m1a0 > SNN_THRESH) ? 1.0f : 0.0f;
        s1a1 = (m1a1 > SNN_THRESH) ? 1.0f : 0.0f;
        s1b0 = (m1b0 > SNN_THRESH) ? 1.0f : 0.0f;
        s1b1 = (m1b1 > SNN_THRESH) ? 1.0f : 0.0f;

        // ---- cur2[b,c] = b2[c] + spk1[b,0]*W2[c,0] + spk1[b,1]*W2[c,1]
        const float c2a0 = fmaf(s1a0, w2_00, fmaf(s1a1, w2_01, b2_0));
        const float c2a1 = fmaf(s1a0, w2_10, fmaf(s1a1, w2_11, b2_1));
        const float c2b0 = fmaf(s1b0, w2_00, fmaf(s1b1, w2_01, b2_0));
        const float c2b1 = fmaf(s1b0, w2_10, fmaf(s1b1, w2_11, b2_1));

        // ---- layer 2
        m2a0 = fmaf(SNN_BETA, m2a0, c2a0) - s2a0;
        m2a1 = fmaf(SNN_BETA, m2a1, c2a1) - s2a1;
        m2b0 = fmaf(SNN_BETA, m2b0, c2b0) - s2b0;
        m2b1 = fmaf(SNN_BETA, m2b1, c2b1) - s2b1;
        s2a0 = (m2a0 > SNN_THRESH) ? 1.0f : 0.0f;
        s2a1 = (m2a1 > SNN_THRESH) ? 1.0f : 0.0f;
        s2b0 = (m2b0 > SNN_THRESH) ? 1.0f : 0.0f;
        s2b1 = (m2b1 > SNN_THRESH) ? 1.0f : 0.0f;

        // ---- record: one b128 NT store each for spk2 and mem2
        v4f sv; sv.x = s2a0; sv.y = s2a1; sv.z = s2b0; sv.w = s2b1;
        v4f mv; mv.x = m2a0; mv.y = m2a1; mv.z = m2b0; mv.w = m2b1;
        __builtin_nontemporal_store(sv, reinterpret_cast<v4f*>(spkPtr));
        __builtin_nontemporal_store(mv, reinterpret_cast<v4f*>(memPtr));
        spkPtr += step;
        memPtr += step;
    }
}

extern "C" void kernel_launch(void* const* d_in, const int* in_sizes, int n_in,
                              void* d_out, int out_size, void* d_ws, size_t ws_size,
                              hipStream_t stream) {
    (void)n_in; (void)out_size; (void)d_ws; (void)ws_size;
    const float* x  = (const float*)d_in[0];
    const float* W1 = (const float*)d_in[1];
    const float* b1 = (const float*)d_in[2];
    const float* W2 = (const float*)d_in[3];
    const float* b2 = (const float*)d_in[4];
    float* out = (float*)d_out;

    const int B = in_sizes[0] / 2;        // x is [B,2]
    const int threads = B / 2;            // 2 batch elements per thread
    const int block = 256;
    const int grid = (threads + block - 1) / block;

    snn_lif_kernel<<<grid, block, 0, stream>>>(x, W1, b1, W2, b2, out, B);
}